// GNN_29961691857024
// MI455X (gfx1250) — compile-verified
//
#include <hip/hip_runtime.h>
#include <hip/hip_bf16.h>
#include <math.h>

#define F_IN  128
#define F_HID 128
#define F_OUT 64

typedef __attribute__((ext_vector_type(2)))  float  v2f;
typedef __attribute__((ext_vector_type(8)))  float  v8f;

// ---------------- degree / normalization ----------------

__global__ void k_fill1(float* __restrict__ p, int n) {
  int i = blockIdx.x * blockDim.x + threadIdx.x;
  if (i < n) p[i] = 1.0f;   // self-loop contributes 1 to every degree
}

__global__ void k_deg(const long long* __restrict__ dst, float* __restrict__ deg, int E) {
  int i = blockIdx.x * blockDim.x + threadIdx.x;
  if (i < E) unsafeAtomicAdd(&deg[(int)dst[i]], 1.0f);
}

__global__ void k_rsqrt(float* __restrict__ p, int n) {
  int i = blockIdx.x * blockDim.x + threadIdx.x;
  if (i < n) p[i] = rsqrtf(p[i]);   // deg >= 1 always (self loops)
}

// ---------------- WMMA GEMM: C[M x NOUT] = act(A[M x K]) @ W[K x NOUT] ----------------
// V_WMMA_F32_16X16X4_F32, fragment layout per CDNA5 ISA 7.12.2:
//   A 16x4 : lane = {M = lane&15}; VGPR j holds K = 2*(lane>>4) + j
//   B 4x16 : lane = {N = lane&15}; VGPR j holds K = 2*(lane>>4) + j  (column of B)
//   C 16x16: VGPR j, lane L -> row = j + 8*(L>>4), col = L&15
//
// Each wave owns one N-tile and a strip of TM M-tiles. The per-lane B fragment
// for the whole K (K/2 floats = 64 VGPRs at K=128) is hoisted into registers,
// so the inner loop is just: load A (b64) -> wmma. Two accumulator chains break
// the C->C dependency between consecutive WMMAs.

template<int K, int NOUT, bool RELU_BIAS, int TM>
__global__ void k_gemm_wmma(const float* __restrict__ A, const float* __restrict__ W,
                            const float* __restrict__ bias, float* __restrict__ C, int M) {
  const int lane    = threadIdx.x & 31;
  const int wave    = (blockIdx.x * blockDim.x + threadIdx.x) >> 5;
  const int tilesN  = NOUT / 16;
  const int mgroups = (M / 16) / TM;
  const int tileN   = wave % tilesN;
  const int mgroup  = wave / tilesN;
  if (mgroup >= mgroups) return;

  const int hi  = lane >> 4;      // 0 or 1 (K-half select)
  const int l15 = lane & 15;
  const int col = tileN * 16 + l15;   // B/D column handled by this lane

  // B fragment for the entire K dimension, resident in VGPRs
  v2f bf[K / 4];
  #pragma unroll
  for (int i = 0; i < K / 4; ++i) {
    const int ka = i * 4 + hi * 2;
    bf[i].x = W[(size_t)ka * NOUT + col];
    bf[i].y = W[(size_t)(ka + 1) * NOUT + col];
  }

  // bias fragment (layer-2 fused relu(agg + b1)), also register-resident
  float bs0[RELU_BIAS ? (K / 4) : 1];
  float bs1[RELU_BIAS ? (K / 4) : 1];
  if (RELU_BIAS) {
    #pragma unroll
    for (int i = 0; i < K / 4; ++i) {
      const int ka = i * 4 + hi * 2;
      bs0[i] = bias[ka];
      bs1[i] = bias[ka + 1];
    }
  }

  for (int t = 0; t < TM; ++t) {
    const int tileM = mgroup * TM + t;
    const float* ap = A + (size_t)(tileM * 16 + l15) * K + hi * 2;
    v8f acc0 = {}, acc1 = {};
    #pragma unroll
    for (int i = 0; i < K / 4; i += 2) {
      v2f a0 = *(const v2f*)(ap + i * 4);
      v2f a1 = *(const v2f*)(ap + i * 4 + 4);
      if (RELU_BIAS) {
        a0.x = fmaxf(a0.x + bs0[i],     0.0f);
        a0.y = fmaxf(a0.y + bs1[i],     0.0f);
        a1.x = fmaxf(a1.x + bs0[i + 1], 0.0f);
        a1.y = fmaxf(a1.y + bs1[i + 1], 0.0f);
      }
      acc0 = __builtin_amdgcn_wmma_f32_16x16x4_f32(false, a0, false, bf[i],
                                                   (short)0, acc0, false, false);
      acc1 = __builtin_amdgcn_wmma_f32_16x16x4_f32(false, a1, false, bf[i + 1],
                                                   (short)0, acc1, false, false);
    }
    const int rbase = tileM * 16 + hi * 8;
    #pragma unroll
    for (int j = 0; j < 8; ++j)
      C[(size_t)(rbase + j) * NOUT + col] = acc0[j] + acc1[j];
  }
}

// ---------------- aggregation ----------------

// agg[i,:] = h[i,:] * dis[i]^2   (self-loop term; also serves as the zero-init)
template<int F>
__global__ void k_selfloop(const float* __restrict__ h, const float* __restrict__ dis,
                           float* __restrict__ agg, int N) {
  long long t = (long long)blockIdx.x * blockDim.x + threadIdx.x;
  if (t >= (long long)N * F) return;
  const int r = (int)(t / F);
  const float w = dis[r];
  agg[t] = h[t] * (w * w);
}

// one edge handled by F/4 lanes; float4 gather + 4 native f32 atomics (L2-resident)
template<int F>
__global__ void k_scatter(const long long* __restrict__ src, const long long* __restrict__ dst,
                          const float* __restrict__ dis, const float* __restrict__ h,
                          float* __restrict__ agg, int E) {
  const int CH = F / 4;
  long long t = (long long)blockIdx.x * blockDim.x + threadIdx.x;
  long long e = t / CH;
  if (e >= E) return;
  const int fc = (int)(t % CH) * 4;
  const int s = (int)src[e];
  const int d = (int)dst[e];
  const float nrm = dis[s] * dis[d];
  const float4 v = *(const float4*)(h + (size_t)s * F + fc);
  float* ap = agg + (size_t)d * F + fc;
  unsafeAtomicAdd(ap + 0, v.x * nrm);
  unsafeAtomicAdd(ap + 1, v.y * nrm);
  unsafeAtomicAdd(ap + 2, v.z * nrm);
  unsafeAtomicAdd(ap + 3, v.w * nrm);
}

// ---------------- final: out = log_softmax(agg2 + b2), one wave per row ----------------

__global__ void k_logsoftmax(const float* __restrict__ agg, const float* __restrict__ b2,
                             float* __restrict__ out, int N) {
  const int wv   = (int)(((long long)blockIdx.x * blockDim.x + threadIdx.x) >> 5);
  const int lane = threadIdx.x & 31;
  if (wv >= N) return;
  const float* rp = agg + (size_t)wv * F_OUT;
  float v0 = rp[lane]      + b2[lane];
  float v1 = rp[lane + 32] + b2[lane + 32];
  float m = fmaxf(v0, v1);
  #pragma unroll
  for (int off = 16; off > 0; off >>= 1) m = fmaxf(m, __shfl_xor(m, off, 32));
  float s = __expf(v0 - m) + __expf(v1 - m);
  #pragma unroll
  for (int off = 16; off > 0; off >>= 1) s += __shfl_xor(s, off, 32);
  const float lse = m + __logf(s);
  float* op = out + (size_t)wv * F_OUT;
  op[lane]      = v0 - lse;
  op[lane + 32] = v1 - lse;
}

// ---------------- launch ----------------

extern "C" void kernel_launch(void* const* d_in, const int* in_sizes, int n_in,
                              void* d_out, int out_size, void* d_ws, size_t ws_size,
                              hipStream_t stream) {
  const float*     x   = (const float*)d_in[0];
  const long long* ei  = (const long long*)d_in[1];
  const float*     W1  = (const float*)d_in[2];
  const float*     b1  = (const float*)d_in[3];
  const float*     W2  = (const float*)d_in[4];
  const float*     b2  = (const float*)d_in[5];
  float*           out = (float*)d_out;

  const int N = in_sizes[0] / F_IN;   // 100000
  const int E = in_sizes[1] / 2;      // 1600000
  const long long* esrc = ei;
  const long long* edst = ei + E;

  char*  ws   = (char*)d_ws;
  float* dis  = (float*)ws;
  size_t off  = ((size_t)N * sizeof(float) + 255) & ~(size_t)255;
  float* bufA = (float*)(ws + off);               // h1, then h2
  float* bufB = bufA + (size_t)N * F_HID;         // agg1, then agg2

  const int B = 256;
  const int TM = 5;                   // 6250 M-tiles = 1250 strips of 5

  // 1. symmetric-norm coefficients
  k_fill1 <<<(N + B - 1) / B, B, 0, stream>>>(dis, N);
  k_deg   <<<(E + B - 1) / B, B, 0, stream>>>(edst, dis, E);
  k_rsqrt <<<(N + B - 1) / B, B, 0, stream>>>(dis, N);

  // 2. layer 1 GEMM: h1 = x @ W1  -> bufA
  {
    const int waves = ((N / 16) / TM) * (F_HID / 16);    // 1250 * 8 = 10000
    k_gemm_wmma<F_IN, F_HID, false, TM><<<(waves * 32 + B - 1) / B, B, 0, stream>>>(x, W1, b1, bufA, N);
  }
  // 3. aggregate layer 1 -> bufB (self-loop init + edge scatter)
  k_selfloop<F_HID><<<(int)(((long long)N * F_HID + B - 1) / B), B, 0, stream>>>(bufA, dis, bufB, N);
  k_scatter <F_HID><<<(int)(((long long)E * (F_HID / 4) + B - 1) / B), B, 0, stream>>>(esrc, edst, dis, bufA, bufB, E);

  // 4. layer 2 GEMM with fused relu(agg1 + b1): h2 = relu(bufB + b1) @ W2 -> bufA
  {
    const int waves = ((N / 16) / TM) * (F_OUT / 16);    // 1250 * 4 = 5000
    k_gemm_wmma<F_HID, F_OUT, true, TM><<<(waves * 32 + B - 1) / B, B, 0, stream>>>(bufB, W2, b1, bufA, N);
  }
  // 5. aggregate layer 2 -> bufB (agg1 is dead now)
  k_selfloop<F_OUT><<<(int)(((long long)N * F_OUT + B - 1) / B), B, 0, stream>>>(bufA, dis, bufB, N);
  k_scatter <F_OUT><<<(int)(((long long)E * (F_OUT / 4) + B - 1) / B), B, 0, stream>>>(esrc, edst, dis, bufA, bufB, E);

  // 6. out = log_softmax(agg2 + b2)
  k_logsoftmax<<<(N * 32 + B - 1) / B, B, 0, stream>>>(bufB, b2, out, N);
}